// MixedScoresSDPA_58351425683887
// MI455X (gfx1250) — compile-verified
//
#include <hip/hip_runtime.h>
#include <math.h>

typedef __attribute__((ext_vector_type(2))) float v2f;
typedef __attribute__((ext_vector_type(8))) float v8f;
typedef __attribute__((ext_vector_type(4))) unsigned int v4u;
typedef __attribute__((ext_vector_type(4))) int v4i;
typedef __attribute__((ext_vector_type(8))) int v8i;

#define B_    4
#define H_    16
#define M_    512
#define N_    512
#define D_    64
#define E_    16
#define NT    32            // keys per chunk == keys per softmax update
#define NCHUNK (N_ / NT)
#define WAVES 4
#define BLOCK (WAVES * 32)

#if defined(__HIP_DEVICE_COMPILE__) && __has_builtin(__builtin_amdgcn_tensor_load_to_lds)
#define HAVE_TDM 1
#else
#define HAVE_TDM 0
#endif

#if HAVE_TDM
// One 2-D TDM tile load (tile_d1 rows x tile_d0 f32 elems) from a
// (tensor_d1 x tensor_d0) tensor with row stride stride0, into LDS.
// D# per CDNA5 ISA ch.8 (group0: count/lds/global/type=2; group1: dsize,
// dims, tile, stride; groups 2/3 zero => 2-D, tile_dim2 unused).
__device__ __forceinline__ void tdm_load_2d(unsigned lds_off, const float* gptr,
                                            unsigned tensor_d0, unsigned tensor_d1,
                                            unsigned stride0,
                                            unsigned tile_d0, unsigned tile_d1)
{
    unsigned long long ga = (unsigned long long)(uintptr_t)gptr;
    v4u g0 = { 1u, lds_off, (unsigned)ga,
               (unsigned)((ga >> 32) & 0x01FFFFFFu) | 0x80000000u };
    v8i g1 = { (int)(2u << 16),
               (int)(tensor_d0 << 16),
               (int)((tensor_d0 >> 16) | (tensor_d1 << 16)),
               (int)((tensor_d1 >> 16) | (tile_d0 << 16)),
               (int)(tile_d1 & 0xffffu),
               (int)stride0,
               0, 0 };
    v4i gz = { 0, 0, 0, 0 };
#if __clang_major__ >= 23
    v8i gz8 = { 0, 0, 0, 0, 0, 0, 0, 0 };
    __builtin_amdgcn_tensor_load_to_lds(g0, g1, gz, gz, gz8, 0);
#else
    __builtin_amdgcn_tensor_load_to_lds(g0, g1, gz, gz, 0);
#endif
}
#endif

__global__ __launch_bounds__(BLOCK) void MixedScoresSDPA_58351425683887_kernel(
    const float* __restrict__ q, const float* __restrict__ k,
    const float* __restrict__ v, const float* __restrict__ dmat,
    const float* __restrict__ w1, const float* __restrict__ b1,
    const float* __restrict__ w2, const float* __restrict__ b2,
    float* __restrict__ out)
{
    __shared__ float lK[2][NT * D_];        // 2 x 8 KB
    __shared__ float lV[2][NT * D_];        // 2 x 8 KB
    __shared__ float lDM[2][64 * NT];       // 2 x 8 KB
    __shared__ float stage[WAVES][16 * NT]; // 8 KB

    const int bid  = blockIdx.x;
    const int mblk = bid % (M_ / 64);
    const int h    = (bid / (M_ / 64)) % H_;
    const int b    = bid / ((M_ / 64) * H_);

    const int tid   = threadIdx.x;
    const int wave  = tid >> 5;
    const int lane  = tid & 31;
    const int laneN = lane & 15;
    const int half  = lane >> 4;

    const int m0   = mblk * 64;
    const int qrow = m0 + wave * 16 + laneN;

    // ---- mixer weights packed along E: natural v2f pairs, no splatting ----
    v2f wa2[E_ / 2], wd2[E_ / 2], wb2v[E_ / 2], ww2[E_ / 2];
    {
        const float* W1 = w1 + h * 2 * E_;
        #pragma unroll
        for (int ep = 0; ep < E_ / 2; ++ep) {
            wa2[ep]  = (*(const v2f*)&W1[2 * ep]) * 0.125f; // fold 1/sqrt(D)
            wd2[ep]  = *(const v2f*)&W1[E_ + 2 * ep];
            wb2v[ep] = *(const v2f*)&b1[h * E_ + 2 * ep];
            ww2[ep]  = *(const v2f*)&w2[h * E_ + 2 * ep];
        }
    }
    const float bias2 = b2[h];

    // ---- preload Q as 16 A-fragments (16x4) ----
    v2f qf[16];
    {
        const float* qbase = q + ((size_t)(b * H_ + h) * M_ + qrow) * D_;
        #pragma unroll
        for (int c = 0; c < 16; ++c)
            qf[c] = *(const v2f*)(qbase + 4 * c + 2 * half);
    }

    // ---- flash state ----
    float mrow[8], lpart[8];
    #pragma unroll
    for (int j = 0; j < 8; ++j) { mrow[j] = -1e30f; lpart[j] = 0.f; }
    v8f ot[4];
    #pragma unroll
    for (int t = 0; t < 4; ++t)
        #pragma unroll
        for (int j = 0; j < 8; ++j) ot[t][j] = 0.f;

    const float* kbase  = k    + (size_t)(b * H_ + h) * N_ * D_;
    const float* vbase  = v    + (size_t)(b * H_ + h) * N_ * D_;
    const float* dmbase = dmat + (size_t)b * M_ * N_;

#if HAVE_TDM
    if (wave == 0) {  // prefetch chunk 0
        tdm_load_2d((unsigned)(size_t)&lK[0][0],  kbase,                    D_, N_, D_, D_, NT);
        tdm_load_2d((unsigned)(size_t)&lV[0][0],  vbase,                    D_, N_, D_, D_, NT);
        tdm_load_2d((unsigned)(size_t)&lDM[0][0], dmbase + (size_t)m0 * N_, N_, M_, N_, NT, 64);
    }
#endif

    for (int c = 0; c < NCHUNK; ++c) {
        const int buf = c & 1;
        const int nc0 = c * NT;

#if HAVE_TDM
        if (wave == 0) __builtin_amdgcn_s_wait_tensorcnt(0);
        __syncthreads();                          // chunk c visible; chunk c-1 reads done
        if (wave == 0 && c + 1 < NCHUNK) {        // prefetch chunk c+1 into other buffer
            const int nn = nc0 + NT;
            tdm_load_2d((unsigned)(size_t)&lK[buf ^ 1][0],  kbase + (size_t)nn * D_, D_, N_, D_, D_, NT);
            tdm_load_2d((unsigned)(size_t)&lV[buf ^ 1][0],  vbase + (size_t)nn * D_, D_, N_, D_, D_, NT);
            tdm_load_2d((unsigned)(size_t)&lDM[buf ^ 1][0], dmbase + (size_t)m0 * N_ + nn, N_, M_, N_, NT, 64);
        }
#else
        __syncthreads();
        #pragma unroll
        for (int it = 0; it < (NT * D_ / 4) / BLOCK; ++it) {
            int idx = tid + it * BLOCK;
            int row = idx >> 4;
            int c4  = (idx & 15) * 4;
            *(float4*)&lK[buf][row * D_ + c4] = *(const float4*)&kbase[(size_t)(nc0 + row) * D_ + c4];
            *(float4*)&lV[buf][row * D_ + c4] = *(const float4*)&vbase[(size_t)(nc0 + row) * D_ + c4];
        }
        #pragma unroll
        for (int it = 0; it < (64 * NT / 4) / BLOCK; ++it) {
            int idx = tid + it * BLOCK;
            int row = idx >> 3;
            int c4  = (idx & 7) * 4;
            *(float4*)&lDM[buf][row * NT + c4] = *(const float4*)&dmbase[(size_t)(m0 + row) * N_ + nc0 + c4];
        }
        __syncthreads();
#endif

        // --- two 16x16 score tiles covering the whole 32-key chunk ---
        v8f sA, sB;
        #pragma unroll
        for (int j = 0; j < 8; ++j) { sA[j] = 0.f; sB[j] = 0.f; }
        #pragma unroll
        for (int cc = 0; cc < 16; ++cc) {
            v2f bfA = *(const v2f*)&lK[buf][laneN * D_ + 4 * cc + 2 * half];
            sA = __builtin_amdgcn_wmma_f32_16x16x4_f32(
                false, qf[cc], false, bfA, (short)0, sA, false, false);
        }
        #pragma unroll
        for (int cc = 0; cc < 16; ++cc) {
            v2f bfB = *(const v2f*)&lK[buf][(16 + laneN) * D_ + 4 * cc + 2 * half];
            sB = __builtin_amdgcn_wmma_f32_16x16x4_f32(
                false, qf[cc], false, bfB, (short)0, sB, false, false);
        }

        // --- mixer MLP, packed along E (weights = natural pairs) ---
        float pA[8], pB[8];
        const v2f zero2 = { 0.f, 0.f };
        #pragma unroll
        for (int j = 0; j < 8; ++j) {
            const int r = (wave * 16 + j + 8 * half) * NT;
            float sa = sA[j];
            float sb = sB[j];
            float da = lDM[buf][r + laneN];
            float db = lDM[buf][r + 16 + laneN];
            v2f accA = zero2, accB = zero2;
            #pragma unroll
            for (int ep = 0; ep < E_ / 2; ++ep) {
                v2f hA = sa * wa2[ep] + (da * wd2[ep] + wb2v[ep]);
                v2f hB = sb * wa2[ep] + (db * wd2[ep] + wb2v[ep]);
                hA = __builtin_elementwise_max(hA, zero2);
                hB = __builtin_elementwise_max(hB, zero2);
                accA = accA + hA * ww2[ep];
                accB = accB + hB * ww2[ep];
            }
            pA[j] = accA.x + accA.y + bias2;
            pB[j] = accB.x + accB.y + bias2;
        }

        // --- one online-softmax update per 32 keys ---
        float alpha[8];
        #pragma unroll
        for (int j = 0; j < 8; ++j) {
            float r = fmaxf(pA[j], pB[j]);
            r = fmaxf(r, __shfl_xor(r, 1));
            r = fmaxf(r, __shfl_xor(r, 2));
            r = fmaxf(r, __shfl_xor(r, 4));
            r = fmaxf(r, __shfl_xor(r, 8));
            float mnew = fmaxf(mrow[j], r);
            alpha[j] = __expf(mrow[j] - mnew);
            mrow[j]  = mnew;
            pA[j] = __expf(pA[j] - mnew);
            pB[j] = __expf(pB[j] - mnew);
            lpart[j] = fmaf(lpart[j], alpha[j], pA[j] + pB[j]);
        }
        // packed alpha rescale of O accumulator (natural v2f pairs)
        #pragma unroll
        for (int jp = 0; jp < 4; ++jp) {
            v2f a2 = { alpha[2 * jp], alpha[2 * jp + 1] };
            #pragma unroll
            for (int t = 0; t < 4; ++t) {
                v2f* otp = (v2f*)&ot[t];
                otp[jp] = otp[jp] * a2;
            }
        }

        // --- relayout both P tiles (C-layout -> A-layout) via per-wave LDS ---
        float* st = stage[wave];
        #pragma unroll
        for (int j = 0; j < 8; ++j) {
            st[(j + 8 * half) * NT + laneN]      = pA[j];
            st[(j + 8 * half) * NT + 16 + laneN] = pB[j];
        }
        v2f pa[8];
        #pragma unroll
        for (int cc = 0; cc < 8; ++cc)
            pa[cc] = *(const v2f*)&st[laneN * NT + 4 * cc + 2 * half];

        // --- O += P * V over all 32 keys of the chunk ---
        #pragma unroll
        for (int t = 0; t < 4; ++t) {
            #pragma unroll
            for (int cc = 0; cc < 8; ++cc) {
                v2f bf;
                bf.x = lV[buf][(4 * cc + 2 * half + 0) * D_ + t * 16 + laneN];
                bf.y = lV[buf][(4 * cc + 2 * half + 1) * D_ + t * 16 + laneN];
                ot[t] = __builtin_amdgcn_wmma_f32_16x16x4_f32(
                    false, pa[cc], false, bf, (short)0, ot[t], false, false);
            }
        }
    }

    // ---- final 1/l normalization + store ----
    #pragma unroll
    for (int j = 0; j < 8; ++j) {
        float l = lpart[j];
        l += __shfl_xor(l, 1);
        l += __shfl_xor(l, 2);
        l += __shfl_xor(l, 4);
        l += __shfl_xor(l, 8);
        float inv = 1.f / l;
        float* orow = out + ((size_t)(b * H_ + h) * M_ + m0 + wave * 16 + j + 8 * half) * D_;
        #pragma unroll
        for (int t = 0; t < 4; ++t)
            orow[t * 16 + laneN] = ot[t][j] * inv;
    }
}

extern "C" void kernel_launch(void* const* d_in, const int* in_sizes, int n_in,
                              void* d_out, int out_size, void* d_ws, size_t ws_size,
                              hipStream_t stream) {
    const float* q    = (const float*)d_in[0];
    const float* k    = (const float*)d_in[1];
    const float* v    = (const float*)d_in[2];
    const float* dmat = (const float*)d_in[3];
    const float* w1   = (const float*)d_in[4];
    const float* b1   = (const float*)d_in[5];
    const float* w2   = (const float*)d_in[6];
    const float* b2   = (const float*)d_in[7];
    float* o = (float*)d_out;

    dim3 grid(B_ * H_ * (M_ / 64));
    dim3 block(BLOCK);
    MixedScoresSDPA_58351425683887_kernel<<<grid, block, 0, stream>>>(
        q, k, v, dmat, w1, b1, w2, b2, o);
}